// AttentionScaleModel_35399120453977
// MI455X (gfx1250) — compile-verified
//
#include <hip/hip_runtime.h>

typedef float v2f __attribute__((ext_vector_type(2)));
typedef float v8f __attribute__((ext_vector_type(8)));

#define S_LEN   2048
#define NB      4
#define NA      4
#define NF      64
#define NL      4
#define KWIN    20      // window radius for scale_idx=4
#define OFF     7       // sum(ATTRIBUTE_NUMS[:3])
#define XROWS   64      // 56 real window rows, padded to 64
#define XPITCH  68      // LDS pitch (floats) to avoid bank conflicts
#define BPITCH  68
#define TILE_S  16

__device__ __forceinline__ v8f wmma_f32(v2f a, v2f b, v8f c) {
    // V_WMMA_F32_16X16X4_F32 : D = A(16x4) * B(4x16) + C(16x16), all f32
    return __builtin_amdgcn_wmma_f32_16x16x4_f32(
        /*neg_a=*/false, a, /*neg_b=*/false, b,
        /*c_mod=*/(short)0, c, /*reuse_a=*/false, /*reuse_b=*/false);
}

// ---------------------------------------------------------------------------
// Prep: per (a,l) fold Wq/Wk into Meff = Wq^T Wk (64x64), ceff = bq*Wk (64),
// and transpose Wt into WtT (64x64, [f][g]).  16 blocks x 256 threads.
// ---------------------------------------------------------------------------
__global__ __launch_bounds__(256)
void attn_prep(const float* __restrict__ Wq, const float* __restrict__ bq,
               const float* __restrict__ Wk, const float* __restrict__ Wt,
               float* __restrict__ meff, float* __restrict__ wtt,
               float* __restrict__ ceff)
{
    const int id = blockIdx.x;          // a*4 + l
    const int a  = id >> 2;
    const int l  = id & 3;
    const size_t wbase = (size_t)((OFF + a) * 5 + l) * (NF * NF);
    const size_t bbase = (size_t)((OFF + a) * 5 + l) * NF;
    const size_t obase = (size_t)id * (NF * NF);
    const int t = threadIdx.x;

    // Meff[f1][f2] = sum_g Wq[g][f1] * Wk[g][f2]; thread t -> f1=t>>2, 16 f2's
    const int f1 = t >> 2;
    const int j0 = (t & 3) * 16;
    float acc[16];
#pragma unroll
    for (int jj = 0; jj < 16; ++jj) acc[jj] = 0.0f;
    for (int g = 0; g < NF; ++g) {
        const float wq = Wq[wbase + (size_t)g * NF + f1];
#pragma unroll
        for (int jj = 0; jj < 16; ++jj)
            acc[jj] += wq * Wk[wbase + (size_t)g * NF + j0 + jj];
    }
#pragma unroll
    for (int jj = 0; jj < 16; ++jj)
        meff[obase + (size_t)f1 * NF + j0 + jj] = acc[jj];

    // WtT[f][g] = Wt[g][f]
#pragma unroll
    for (int i = 0; i < 16; ++i) {
        const int e = t * 16 + i;
        const int f = e >> 6, g = e & 63;
        wtt[obase + e] = Wt[wbase + (size_t)g * NF + f];
    }

    // ceff[f2] = sum_g bq[g] * Wk[g][f2]
    if (t < NF) {
        float cc = 0.0f;
        for (int g = 0; g < NF; ++g)
            cc += bq[bbase + g] * Wk[wbase + (size_t)g * NF + t];
        ceff[(size_t)id * NF + t] = cc;
    }
}

// ---------------------------------------------------------------------------
// Main fused windowed-attention kernel.
// grid = (S/16, b*a), block = 128 (4 wave32, wave w handles level l=w)
// ---------------------------------------------------------------------------
__global__ __launch_bounds__(128)
void attn_main(const float* __restrict__ x,     // (b,s,a,f)
               const float* __restrict__ bt,    // (14,5,64)
               const float* __restrict__ meff,  // (16,64,64)
               const float* __restrict__ wtt,   // (16,64,64)
               const float* __restrict__ ceff,  // (16,64)
               float* __restrict__ out)         // (b,s,a,l,g)
{
    __shared__ float Xext[XROWS * XPITCH];
    __shared__ float wbuf[NL * TILE_S * BPITCH];
    __shared__ float rowmask[XROWS];

    const int tid  = threadIdx.x;
    const int lane = tid & 31;
    const int l    = tid >> 5;
    const int s0   = blockIdx.x * TILE_S;
    const int b    = blockIdx.y >> 2;
    const int a    = blockIdx.y & 3;

    // ---- load extended window tile (rows s0-20 .. s0+43), zero padded ----
    for (int e = tid; e < XROWS * NF; e += 128) {
        const int j = e >> 6, f = e & 63;
        const int sp = s0 + j - KWIN;
        float v = 0.0f;
        if (sp >= 0 && sp < S_LEN)
            v = x[(((size_t)b * S_LEN + sp) * NA + a) * NF + f];
        Xext[j * XPITCH + f] = v;
    }
    __syncthreads();
    if (tid < XROWS) {
        int allz = 1;
        for (int f = 0; f < NF; ++f)
            allz &= (Xext[tid * XPITCH + f] == 0.0f) ? 1 : 0;
        rowmask[tid] = allz ? 1.0f : 0.0f;
    }
    __syncthreads();

    float* wb = &wbuf[l * TILE_S * BPITCH];
    const int half = (lane < 16) ? 0 : 2;   // K sub-offset for A/B fragments
    const int nl_  = lane & 15;             // M (for A) / N (for B,C) in-lane idx

    const float* M  = meff + (size_t)(a * NL + l) * (NF * NF);
    const float* WT = wtt  + (size_t)(a * NL + l) * (NF * NF);
    const float* CE = ceff + (size_t)(a * NL + l) * NF;

    // ================= GEMM1: QK = Xtile @ Meff + ceff =================
    v8f qk[4];
#pragma unroll
    for (int t = 0; t < 4; ++t) {
        const float ci = CE[t * 16 + nl_];
        v8f c = {ci, ci, ci, ci, ci, ci, ci, ci};
        for (int kk = 0; kk < 16; ++kk) {
            const int k0 = kk * 4 + half;
            v2f af, bf;
            af.x = Xext[(KWIN + nl_) * XPITCH + k0];
            af.y = Xext[(KWIN + nl_) * XPITCH + k0 + 1];
            bf.x = M[(size_t)k0 * NF + t * 16 + nl_];
            bf.y = M[(size_t)(k0 + 1) * NF + t * 16 + nl_];
            c = wmma_f32(af, bf, c);
        }
        qk[t] = c;
    }
#pragma unroll
    for (int t = 0; t < 4; ++t)
#pragma unroll
        for (int j = 0; j < 8; ++j) {
            const int r = (lane < 16) ? j : j + 8;
            wb[r * BPITCH + t * 16 + nl_] = qk[t][j];
        }

    // ================= GEMM2: attn = QK @ Xext^T  (16 x 64 band) =======
    v8f at[4];
#pragma unroll
    for (int t = 0; t < 4; ++t) {
        v8f c = {0, 0, 0, 0, 0, 0, 0, 0};
        for (int kk = 0; kk < 16; ++kk) {
            const int k0 = kk * 4 + half;
            v2f af, bf;
            af.x = wb[nl_ * BPITCH + k0];
            af.y = wb[nl_ * BPITCH + k0 + 1];
            bf.x = Xext[(t * 16 + nl_) * XPITCH + k0];      // B[k][n] = Xext[n][k]
            bf.y = Xext[(t * 16 + nl_) * XPITCH + k0 + 1];
            c = wmma_f32(af, bf, c);
        }
        at[t] = c;
    }
#pragma unroll
    for (int t = 0; t < 4; ++t)
#pragma unroll
        for (int j = 0; j < 8; ++j) {
            const int r = (lane < 16) ? j : j + 8;
            wb[r * BPITCH + t * 16 + nl_] = at[t][j];
        }

    // ================= softmax over the 41-wide band (in place) ========
    if (lane < 16) {
        const int r = lane;
        float mx = -3.0e38f;
        for (int i = 0; i <= 2 * KWIN; ++i) {
            const int j = r + i;
            const float v = (rowmask[j] != 0.0f) ? -1.0e9f
                                                 : wb[r * BPITCH + j] * 0.125f;
            mx = fmaxf(mx, v);
        }
        float sum = 0.0f;
        for (int i = 0; i <= 2 * KWIN; ++i) {
            const int j = r + i;
            const float v = (rowmask[j] != 0.0f) ? -1.0e9f
                                                 : wb[r * BPITCH + j] * 0.125f;
            const float e = __expf(v - mx);
            sum += e;
            wb[r * BPITCH + j] = e;
        }
        const float inv = 1.0f / sum;
        for (int j = 0; j < XROWS; ++j) {
            const float wv = (j >= r && j <= r + 2 * KWIN)
                                 ? wb[r * BPITCH + j] * inv : 0.0f;
            wb[r * BPITCH + j] = wv;   // banded weight matrix, zero outside
        }
    }

    // ================= GEMM3: ctx = Wgt(16x64 band) @ Xext =============
    v8f cx[4];
#pragma unroll
    for (int t = 0; t < 4; ++t) {
        v8f c = {0, 0, 0, 0, 0, 0, 0, 0};
        for (int kk = 0; kk < 16; ++kk) {
            const int k0 = kk * 4 + half;
            v2f af, bf;
            af.x = wb[nl_ * BPITCH + k0];
            af.y = wb[nl_ * BPITCH + k0 + 1];
            bf.x = Xext[k0 * XPITCH + t * 16 + nl_];        // B[k][n] = Xext[k][n]
            bf.y = Xext[(k0 + 1) * XPITCH + t * 16 + nl_];
            c = wmma_f32(af, bf, c);
        }
        cx[t] = c;
    }
#pragma unroll
    for (int t = 0; t < 4; ++t)
#pragma unroll
        for (int j = 0; j < 8; ++j) {
            const int r = (lane < 16) ? j : j + 8;
            wb[r * BPITCH + t * 16 + nl_] = cx[t][j];
        }

    // ================= GEMM4: out = ctx @ WtT + bt =====================
#pragma unroll
    for (int t = 0; t < 4; ++t) {
        const float bias = bt[(size_t)((OFF + a) * 5 + l) * NF + t * 16 + nl_];
        v8f c = {bias, bias, bias, bias, bias, bias, bias, bias};
        for (int kk = 0; kk < 16; ++kk) {
            const int k0 = kk * 4 + half;
            v2f af, bf;
            af.x = wb[nl_ * BPITCH + k0];
            af.y = wb[nl_ * BPITCH + k0 + 1];
            bf.x = WT[(size_t)k0 * NF + t * 16 + nl_];
            bf.y = WT[(size_t)(k0 + 1) * NF + t * 16 + nl_];
            c = wmma_f32(af, bf, c);
        }
#pragma unroll
        for (int j = 0; j < 8; ++j) {
            const int r = (lane < 16) ? j : j + 8;
            out[((((size_t)b * S_LEN + s0 + r) * NA + a) * NL + l) * NF
                + t * 16 + nl_] = c[j];
        }
    }
}

extern "C" void kernel_launch(void* const* d_in, const int* in_sizes, int n_in,
                              void* d_out, int out_size, void* d_ws, size_t ws_size,
                              hipStream_t stream) {
    const float* x  = (const float*)d_in[0];
    const float* Wq = (const float*)d_in[1];
    const float* bq = (const float*)d_in[2];
    const float* Wk = (const float*)d_in[3];
    // bk (d_in[4]) contributes only a per-row additive constant over the
    // window axis -> exactly cancelled by softmax, so it is unused.
    const float* Wt = (const float*)d_in[5];
    const float* bt = (const float*)d_in[6];

    float* meff = (float*)d_ws;                 // 16 * 64 * 64
    float* wtt  = meff + 16 * NF * NF;          // 16 * 64 * 64
    float* ceff = wtt  + 16 * NF * NF;          // 16 * 64

    attn_prep<<<16, 256, 0, stream>>>(Wq, bq, Wk, Wt, meff, wtt, ceff);

    dim3 grid(S_LEN / TILE_S, NB * NA);
    attn_main<<<grid, 128, 0, stream>>>(x, bt, meff, wtt, ceff, (float*)d_out);
}